// SpatialCrossAttention_21741124453154
// MI455X (gfx1250) — compile-verified
//
#include <hip/hip_runtime.h>

#define BEV_H 100
#define BEV_W 100
#define BEV_Z 4
#define NUM_CAMS 6
#define EMBED 256
#define HF 32
#define WF 88
#define NQ (BEV_H * BEV_W)        // 10000
#define NPIX (HF * WF)            // 2816
#define NJ (NUM_CAMS * BEV_Z)     // 24
#define EPS 1e-5f

typedef __attribute__((ext_vector_type(2))) float v2f;
typedef __attribute__((ext_vector_type(8))) float v8f;
typedef __attribute__((ext_vector_type(4))) unsigned int v4u;
typedef __attribute__((ext_vector_type(8))) int v8i;
typedef __attribute__((ext_vector_type(4))) int v4i;

// ---------------------------------------------------------------------------
// Kernel 1: transpose img_feats (cam, C, H, W) -> (cam, H*W, C) so that one
// bilinear tap reads 256 contiguous floats (1KB vector) instead of a 11KB-
// strided gather. 17.3MB, stays resident in the 192MB L2.
// ---------------------------------------------------------------------------
__global__ void transpose_feats(const float* __restrict__ in,
                                float* __restrict__ out) {
    int tid = blockIdx.x * blockDim.x + threadIdx.x;
    const int total = NUM_CAMS * NPIX * EMBED;
    if (tid >= total) return;
    int c   = tid & (EMBED - 1);
    int pg  = tid >> 8;             // cam*NPIX + pix
    int cam = pg / NPIX;
    int pix = pg - cam * NPIX;
    out[tid] = in[(cam * EMBED + c) * NPIX + pix];
}

// ---------------------------------------------------------------------------
// Kernel 2: per-query projection + softmax + weighted bilinear aggregation.
// One block per query (256 threads = 8 wave32s, one channel per thread).
// ---------------------------------------------------------------------------
__global__ void sample_agg(const float* __restrict__ query,
                           const float* __restrict__ featsT,
                           const float* __restrict__ l2i,
                           const float* __restrict__ attn_w,
                           const float* __restrict__ attn_b,
                           float* __restrict__ agg) {
    __shared__ float qlds[EMBED];
    __shared__ float logits[NJ];
    __shared__ float wt[NJ * 4];
    __shared__ int   pidx[NJ * 4];

    const int q = blockIdx.x;
    const int t = threadIdx.x;

    qlds[t] = query[q * EMBED + t];
    __syncthreads();

    if (t < NJ) {
        // ---- attention logit for (cam,z) pair t ----
        float lg = attn_b[t];
        for (int c = 0; c < EMBED; ++c) lg += qlds[c] * attn_w[c * NJ + t];
        logits[t] = lg;

        // ---- camera projection ----
        int cam = t >> 2;
        int z   = t & 3;
        int hq  = q / BEV_W;
        int wq  = q - hq * BEV_W;
        float rx = (wq + 0.5f) * (1.0f / BEV_W);
        float ry = (hq + 0.5f) * (1.0f / BEV_H);
        float rz = (z  + 0.5f) * (1.0f / BEV_Z);
        float X  = rx * 102.4f - 51.2f;
        float Y  = ry * 102.4f - 51.2f;
        float Zw = rz * 8.0f   - 5.0f;
        const float* M = l2i + cam * 16;
        float cx = M[0] * X + M[1] * Y + M[2]  * Zw + M[3];
        float cy = M[4] * X + M[5] * Y + M[6]  * Zw + M[7];
        float cd = M[8] * X + M[9] * Y + M[10] * Zw + M[11];
        float d  = fmaxf(cd, EPS);
        float xi = cx / d;
        float yi = cy / d;
        float gx = (xi * (1.0f / (WF - 1)) - 0.5f) * 2.0f;
        float gy = (yi * (1.0f / (HF - 1)) - 0.5f) * 2.0f;
        bool valid = (cd > EPS) && (gx > -1.0f) && (gx < 1.0f)
                                && (gy > -1.0f) && (gy < 1.0f);
        // grid_sample mapping (align_corners=False)
        float xs = ((gx + 1.0f) * WF - 1.0f) * 0.5f;
        float ys = ((gy + 1.0f) * HF - 1.0f) * 0.5f;
        float x0 = floorf(xs), y0 = floorf(ys);
        float wx1 = xs - x0, wy1 = ys - y0;
        int ix0 = (int)x0, iy0 = (int)y0;
        float tw[4] = { (1.0f - wx1) * (1.0f - wy1), wx1 * (1.0f - wy1),
                        (1.0f - wx1) * wy1,          wx1 * wy1 };
        int tx[4] = { ix0, ix0 + 1, ix0,     ix0 + 1 };
        int ty[4] = { iy0, iy0,     iy0 + 1, iy0 + 1 };
        #pragma unroll
        for (int k = 0; k < 4; ++k) {
            bool tv = valid && (tx[k] >= 0) && (tx[k] < WF)
                            && (ty[k] >= 0) && (ty[k] < HF);
            int xc = min(max(tx[k], 0), WF - 1);
            int yc = min(max(ty[k], 0), HF - 1);
            wt[t * 4 + k]   = tv ? tw[k] : 0.0f;     // NaN-safe: invalid -> 0
            pidx[t * 4 + k] = cam * NPIX + yc * WF + xc;
        }
    }
    __syncthreads();

    if (t == 0) {   // softmax over 24 values: cheap serial pass
        float m = logits[0];
        for (int j = 1; j < NJ; ++j) m = fmaxf(m, logits[j]);
        float s = 0.0f;
        for (int j = 0; j < NJ; ++j) { float e = expf(logits[j] - m); logits[j] = e; s += e; }
        float inv = 1.0f / s;
        for (int j = 0; j < NJ; ++j) logits[j] *= inv;
    }
    __syncthreads();
    if (t < NJ * 4) wt[t] *= logits[t >> 2];
    __syncthreads();

    // ---- weighted gather: 96 taps, weight is block-uniform so the skip
    //      branch is divergence-free and drops loads for invalid points ----
    float acc = 0.0f;
    for (int i = 0; i < NJ * 4; ++i) {
        float w = wt[i];
        if (w != 0.0f) acc += w * featsT[(size_t)pidx[i] * EMBED + t];
    }
    agg[q * EMBED + t] = acc;
}

// ---------------------------------------------------------------------------
// Kernel 3: out = agg @ out_w + out_b + query via V_WMMA_F32_16X16X4_F32.
// 625 blocks x 512 threads (16 wave32s). Block owns one 16-query M tile,
// each wave owns one 16-wide N tile.
//
// A tile (16x256 f32 = 16KB) is staged into LDS by the Tensor Data Mover:
// one tensor_load_to_lds issued by wave 0, TDM pad feature (1 DWORD pad per
// 256 DWORDs) realizes the pitch-257 bank-conflict-free layout in hardware.
// ---------------------------------------------------------------------------
__global__ void out_gemm(const float* __restrict__ agg,
                         const float* __restrict__ out_w,
                         const float* __restrict__ out_b,
                         const float* __restrict__ query,
                         float* __restrict__ out) {
    __shared__ float ldsA[16 * 257];

    const int q0 = blockIdx.x * 16;
    const int t  = threadIdx.x;          // 0..511

    if (t < 32) {
        // ---- Tensor DMA descriptor (D#), 2-D tile: 256 x 16 f32 ----
        unsigned long long ga =
            (unsigned long long)(const void*)(agg + (size_t)q0 * EMBED);
        unsigned int lds_off = (unsigned int)(size_t)(void*)ldsA; // LDS offset = addr[31:0]

        v4u g0;
        g0[0] = 1u;                                   // count=1, is_restore=0, no gather
        g0[1] = lds_off;                              // lds_addr (bytes)
        g0[2] = (unsigned int)ga;                     // global_addr[31:0]
        g0[3] = (unsigned int)(ga >> 32) | (2u << 30);// global_addr[56:32] | type=2

        v8i g1;
        // data_size=2 (4B) | pad_enable | pad_interval=7 (256 DWORDs), pad_amount=0 (1 DWORD)
        g1[0] = (2 << 16) | (1 << 20) | (7 << 22);
        g1[1] = (int)(EMBED << 16);                   // tensor_dim0 = 256 (low 16 in [31:16])
        g1[2] = (int)(16 << 16);                      // tensor_dim0 hi=0 | tensor_dim1 = 16
        g1[3] = (int)(EMBED << 16);                   // tensor_dim1 hi=0 | tile_dim0 = 256
        g1[4] = 16;                                   // tile_dim1 = 16, tile_dim2 = 0
        g1[5] = EMBED;                                // tensor_dim0_stride = 256 (low 32)
        g1[6] = 0;                                    // stride0 hi | tensor_dim1_stride lo
        g1[7] = 0;
        v4i gz = (v4i)0;                              // groups 2/3 unused (2-D tensor)

#if defined(__clang_major__) && (__clang_major__ >= 23)
        __builtin_amdgcn_tensor_load_to_lds(g0, g1, gz, gz, (v8i)0, 0);
#else
        __builtin_amdgcn_tensor_load_to_lds(g0, g1, gz, gz, 0);
#endif
        __builtin_amdgcn_s_wait_tensorcnt(0);         // s_wait_tensorcnt 0
    }
    __syncthreads();

    const int wave = t >> 5;             // nTile 0..15
    const int lane = t & 31;
    const int mA   = lane & 15;          // A-frag: M = lane%16 in both halves
    const int n    = lane & 15;
    const int klo  = (lane < 16) ? 0 : 2;  // lane halves hold K 0/1 vs 2/3
    const float* B = out_w + wave * 16;

    v8f acc = 0.0f;
    #pragma unroll 8
    for (int kk = 0; kk < EMBED; kk += 4) {
        v2f a, b;
        a.x = ldsA[mA * 257 + kk + klo];
        a.y = ldsA[mA * 257 + kk + klo + 1];
        b.x = B[(kk + klo)     * EMBED + n];
        b.y = B[(kk + klo + 1) * EMBED + n];
        // 8-arg form: (neg_a, A, neg_b, B, c_mod, C, reuse_a, reuse_b)
        acc = __builtin_amdgcn_wmma_f32_16x16x4_f32(
            false, a, false, b, (short)0, acc, false, false);
    }

    const int nn   = wave * 16 + n;
    const float bv = out_b[nn];
    #pragma unroll
    for (int r = 0; r < 8; ++r) {
        int m = r + ((lane < 16) ? 0 : 8);   // C/D layout: row = vgpr + 8*hi
        int q = q0 + m;
        out[q * EMBED + nn] = acc[r] + bv + query[q * EMBED + nn];
    }
}

// ---------------------------------------------------------------------------
extern "C" void kernel_launch(void* const* d_in, const int* in_sizes, int n_in,
                              void* d_out, int out_size, void* d_ws, size_t ws_size,
                              hipStream_t stream) {
    const float* query     = (const float*)d_in[0];
    const float* img_feats = (const float*)d_in[1];
    const float* lidar2img = (const float*)d_in[2];
    const float* attn_w    = (const float*)d_in[3];
    const float* attn_b    = (const float*)d_in[4];
    const float* out_w     = (const float*)d_in[5];
    const float* out_b     = (const float*)d_in[6];
    float* out = (float*)d_out;

    // workspace layout: featsT (17.3MB) | agg (10.24MB)
    float* featsT = (float*)d_ws;
    float* agg    = (float*)((char*)d_ws + (size_t)NUM_CAMS * NPIX * EMBED * sizeof(float));

    const int totalT = NUM_CAMS * NPIX * EMBED;
    transpose_feats<<<(totalT + 255) / 256, 256, 0, stream>>>(img_feats, featsT);
    sample_agg<<<NQ, 256, 0, stream>>>(query, featsT, lidar2img, attn_w, attn_b, agg);
    out_gemm<<<NQ / 16, 512, 0, stream>>>(agg, out_w, out_b, query, out);
}